// AdaptiveGraphAttention_43679817401134
// MI455X (gfx1250) — compile-verified
//
#include <hip/hip_runtime.h>
#include <math.h>

#define BB 8
#define SS 128
#define NN 129
#define DD 256
#define HH 8
#define HD 32

typedef float v2f __attribute__((ext_vector_type(2)));
typedef float v8f __attribute__((ext_vector_type(8)));

__device__ __forceinline__ float sigmoidf(float x) {
    return 1.0f / (1.0f + __expf(-x));
}

// Block-wide sum over 256 threads, result broadcast to all threads.
__device__ __forceinline__ float blockReduceSum(float v, float* s, int tid) {
    for (int o = 16; o > 0; o >>= 1) v += __shfl_xor(v, o, 32);
    __syncthreads();                  // protect s from previous use
    if ((tid & 31) == 0) s[tid >> 5] = v;
    __syncthreads();
    return s[0] + s[1] + s[2] + s[3] + s[4] + s[5] + s[6] + s[7];
}

// ---------------------------------------------------------------------------
// C[M x 256] = A[M x 256] @ W[256 x 256] (+bias), row-major, f32 WMMA.
// One wave (32 threads) per 16x16 output tile; K-loop of 64 wmma 16x16x4.
// A layout (16x4): lane<16 holds row l, K=k0..k0+1; lane>=16 row l-16, K=k0+2..3
// B layout (4x16): VGPR0 = B[k0+2*half][n], VGPR1 = B[k0+2*half+1][n], n=lane&15
// C/D layout: VGPR r -> (M = r + 8*half, N = lane&15)
// ---------------------------------------------------------------------------
__global__ void gemm_wmma(const float* __restrict__ A, const float* __restrict__ W,
                          const float* __restrict__ bias, float* __restrict__ C, int M) {
    const int tm = blockIdx.x >> 4;   // 16 N-tiles (DD/16)
    const int tn = blockIdx.x & 15;
    const int lane = threadIdx.x;
    const int half = lane >> 4;
    const int l15  = lane & 15;

    int arow = tm * 16 + l15;
    if (arow >= M) arow = M - 1;      // clamp loads; stores masked below
    const float* Ap = A + (size_t)arow * DD + half * 2;
    const float* Wp = W + (size_t)(half * 2) * DD + tn * 16 + l15;

    v8f c = {0.f, 0.f, 0.f, 0.f, 0.f, 0.f, 0.f, 0.f};
    for (int k0 = 0; k0 < DD; k0 += 4) {
        v2f a = { Ap[k0], Ap[k0 + 1] };
        v2f b = { Wp[(size_t)k0 * DD], Wp[(size_t)(k0 + 1) * DD] };
        c = __builtin_amdgcn_wmma_f32_16x16x4_f32(false, a, false, b,
                                                  (short)0, c, false, false);
    }

    const int col = tn * 16 + l15;
    const float bv = bias ? bias[col] : 0.0f;
    for (int r = 0; r < 8; ++r) {
        int row = tm * 16 + r + half * 8;
        if (row < M) C[(size_t)row * DD + col] = c[r] + bv;
    }
}

// ---------------------------------------------------------------------------
// Row inverse norms of d (B,S,D) and var = nv[:,1:,:]. One wave per row.
// ---------------------------------------------------------------------------
__global__ void norms_kernel(const float* __restrict__ d, const float* __restrict__ nv,
                             float* __restrict__ invd, float* __restrict__ invv) {
    const int wid = threadIdx.x >> 5, lane = threadIdx.x & 31;
    const int row = blockIdx.x * 8 + wid;            // 0..B*S-1
    const int b = row >> 7, s = row & 127;
    const float* dp = d + (size_t)row * DD;
    const float* vp = nv + ((size_t)b * NN + 1 + s) * DD;
    float s1 = 0.f, s2 = 0.f;
    for (int t = 0; t < 8; ++t) {
        float x = dp[lane + 32 * t]; s1 += x * x;
        float y = vp[lane + 32 * t]; s2 += y * y;
    }
    for (int o = 16; o > 0; o >>= 1) {
        s1 += __shfl_xor(s1, o, 32);
        s2 += __shfl_xor(s2, o, 32);
    }
    if (lane == 0) { invd[row] = rsqrtf(s1); invv[row] = rsqrtf(s2); }
}

// ---------------------------------------------------------------------------
// Fused adjacency: dn@dn^T and vn@vn^T via two WMMA accumulator chains,
// then sigmoid/threshold epilogue -> hard na interior (i,j in 1..S).
// One wave per 16x16 tile; grid (64 tiles, 8 batches).
// ---------------------------------------------------------------------------
__global__ void adj_wmma(const float* __restrict__ d, const float* __restrict__ nv,
                         const float* __restrict__ invd, const float* __restrict__ invv,
                         const float* __restrict__ topo, float* __restrict__ na) {
    const int tm = blockIdx.x >> 3;   // 8 tiles per dim (S=128)
    const int tn = blockIdx.x & 7;
    const int b  = blockIdx.y;
    const int lane = threadIdx.x, half = lane >> 4, l15 = lane & 15;

    const float* dB = d + (size_t)b * SS * DD;
    const float* vB = nv + ((size_t)b * NN + 1) * DD;   // var = nv[:,1:,:]
    const float* Ad = dB + (size_t)(tm * 16 + l15) * DD + half * 2;
    const float* Bd = dB + (size_t)(tn * 16 + l15) * DD + half * 2;  // B = d^T tile
    const float* Av = vB + (size_t)(tm * 16 + l15) * DD + half * 2;
    const float* Bv = vB + (size_t)(tn * 16 + l15) * DD + half * 2;

    v8f cg = {0.f,0.f,0.f,0.f,0.f,0.f,0.f,0.f};
    v8f cs = {0.f,0.f,0.f,0.f,0.f,0.f,0.f,0.f};
    for (int k0 = 0; k0 < DD; k0 += 4) {
        v2f a1 = { Ad[k0], Ad[k0 + 1] };
        v2f b1 = { Bd[k0], Bd[k0 + 1] };
        cg = __builtin_amdgcn_wmma_f32_16x16x4_f32(false, a1, false, b1,
                                                   (short)0, cg, false, false);
        v2f a2 = { Av[k0], Av[k0 + 1] };
        v2f b2 = { Bv[k0], Bv[k0 + 1] };
        cs = __builtin_amdgcn_wmma_f32_16x16x4_f32(false, a2, false, b2,
                                                   (short)0, cs, false, false);
    }

    const float tb = topo[0];
    for (int r = 0; r < 8; ++r) {
        int i = tm * 16 + r + half * 8;   // 0..S-1
        int j = tn * 16 + l15;
        float g  = sigmoidf(cg[r] * invd[b * SS + i] * invd[b * SS + j] + tb);
        float ss = cs[r] * invv[b * SS + i] * invv[b * SS + j];
        float m  = (i == j) ? 0.0f : g * ss;
        float adj = sigmoidf(m);
        na[((size_t)b * NN + (i + 1)) * NN + (j + 1)] = (adj > 0.5f) ? 1.0f : 0.0f;
    }
}

// na boundary: na[b,:,0] = 0 ; na[b,0,j>=1] = 1 ; na[b,0,0] = 0
__global__ void na_boundary(float* __restrict__ na) {
    int idx = blockIdx.x * blockDim.x + threadIdx.x;
    if (idx >= BB * NN) return;
    int b = idx / NN, t = idx % NN;
    na[((size_t)b * NN + t) * NN + 0] = 0.0f;
    na[((size_t)b * NN + 0) * NN + t] = (t == 0) ? 0.0f : 1.0f;
}

// U[d,h] = sum_c W_e2[d, h*hd+c]*we[c] ; c0[h] = b_e2_h . we
__global__ void precompute_u(const float* __restrict__ W_e2, const float* __restrict__ b_e2,
                             const float* __restrict__ w_attn,
                             float* __restrict__ U, float* __restrict__ c0) {
    const int dd = threadIdx.x;   // 0..255
    for (int h = 0; h < HH; ++h) {
        float acc = 0.f;
        for (int c = 0; c < HD; ++c)
            acc += W_e2[(size_t)dd * DD + h * HD + c] * w_attn[2 * HD + c];
        U[dd * HH + h] = acc;
    }
    if (dd < HH) {
        float acc = 0.f;
        for (int c = 0; c < HD; ++c) acc += b_e2[dd * HD + c] * w_attn[2 * HD + c];
        c0[dd] = acc;
    }
}

// sq[bn,h] = q_head . wq ; sk[bn,h] = k_head . wk
__global__ void sqsk_kernel(const float* __restrict__ q, const float* __restrict__ k,
                            const float* __restrict__ w_attn,
                            float* __restrict__ sq, float* __restrict__ sk) {
    int idx = blockIdx.x * blockDim.x + threadIdx.x;
    if (idx >= BB * NN * HH) return;
    int h = idx % HH, bn = idx / HH;
    float aq = 0.f, ak = 0.f;
    for (int c = 0; c < HD; ++c) {
        aq += q[(size_t)bn * DD + h * HD + c] * w_attn[c];
        ak += k[(size_t)bn * DD + h * HD + c] * w_attn[HD + c];
    }
    sq[idx] = aq; sk[idx] = ak;
}

// ---------------------------------------------------------------------------
// Fused scores: for row (b,i), loop j: LN(pre)+ReLU, 8 head dots with U,
// add sq/sk/bias and mask, write raw scores into attn output region.
// na is block-uniform per (i,j) -> skip LN work when edge is masked.
// ---------------------------------------------------------------------------
__global__ void scores_kernel(const float* __restrict__ a, const float* __restrict__ bm,
                              const float* __restrict__ b_e1, const float* __restrict__ ln_g,
                              const float* __restrict__ ln_b, const float* __restrict__ U,
                              const float* __restrict__ c0, const float* __restrict__ na,
                              const float* __restrict__ sq, const float* __restrict__ sk,
                              const float* __restrict__ b_attn, float* __restrict__ attn) {
    __shared__ float red[8];
    const int bi = blockIdx.x;           // b*N + i
    const int b = bi / NN, i = bi % NN;
    const int tid = threadIdx.x;         // feature index d

    float av = 0.f;
    if (i > 0) av = a[((size_t)b * SS + i - 1) * DD + tid];
    const float be1 = b_e1[tid], g = ln_g[tid], beta = ln_b[tid];
    float u[HH];
    for (int h = 0; h < HH; ++h) u[h] = U[tid * HH + h];
    const float batt = b_attn[0];

    for (int j = 0; j < NN; ++j) {
        const float naval = na[((size_t)b * NN + i) * NN + j];   // block-uniform
        float dots[HH];
        if (naval != 0.0f) {
            float pa, pb;
            if (i > 0 && j > 0)      { pa = av; pb = bm[((size_t)b * SS + j - 1) * DD + tid]; }
            else if (i == 0 && j > 0){ pa = a [((size_t)b * SS + j - 1) * DD + tid];
                                       pb = bm[((size_t)b * SS + j - 1) * DD + tid]; }
            else if (i > 0)          { pa = av; pb = bm[((size_t)b * SS + i - 1) * DD + tid]; }
            else                     { pa = 0.f; pb = 0.f; }
            float pre = pa + pb + be1;
            float mu = blockReduceSum(pre, red, tid) * (1.0f / DD);
            float dv = pre - mu;
            float vr = blockReduceSum(dv * dv, red, tid) * (1.0f / DD);
            float x = dv * rsqrtf(vr + 1e-5f) * g + beta;
            x = fmaxf(x, 0.0f);
            for (int h = 0; h < HH; ++h)
                dots[h] = blockReduceSum(x * u[h], red, tid);
        }
        if (tid < HH) {
            const int h = tid;
            float e = (naval != 0.0f) ? naval * (dots[h] + c0[h]) : 0.0f;
            float sc = sq[bi * HH + h] + sk[(b * NN + j) * HH + h] + e + batt
                     + ((naval == 0.0f) ? -1.0e9f : 0.0f);
            attn[(((size_t)b * HH + h) * NN + i) * NN + j] = sc;
        }
        __syncthreads();
    }
}

// In-place softmax over j (N=129) per (b,h,i) row; one wave per row.
__global__ void softmax_kernel(float* __restrict__ attn) {
    const int wid = threadIdx.x >> 5, lane = threadIdx.x & 31;
    const int row = blockIdx.x * 8 + wid;          // (b*H+h)*N + i
    if (row >= BB * HH * NN) return;
    float* p = attn + (size_t)row * NN;
    float vals[5];
    float mx = -1.0e30f;
    for (int t = 0; t < 5; ++t) {
        int j = lane + 32 * t;
        vals[t] = (j < NN) ? p[j] : -1.0e30f;
        mx = fmaxf(mx, vals[t]);
    }
    for (int o = 16; o > 0; o >>= 1) mx = fmaxf(mx, __shfl_xor(mx, o, 32));
    float sum = 0.f;
    for (int t = 0; t < 5; ++t) {
        int j = lane + 32 * t;
        if (j < NN) { vals[t] = __expf(vals[t] - mx); sum += vals[t]; }
    }
    for (int o = 16; o > 0; o >>= 1) sum += __shfl_xor(sum, o, 32);
    const float inv = 1.0f / sum;
    for (int t = 0; t < 5; ++t) {
        int j = lane + 32 * t;
        if (j < NN) p[j] = vals[t] * inv;
    }
}

// ctx[b,i,h*hd+dd] = sum_j attn[b,h,i,j] * v[b,j,h*hd+dd]
__global__ void ctx_kernel(const float* __restrict__ attn, const float* __restrict__ v,
                           float* __restrict__ ctx) {
    const int bi = blockIdx.x;           // b*N + i
    const int b = bi / NN, i = bi % NN;
    const int tid = threadIdx.x;         // h*HD + dd
    const int h = tid >> 5;
    const float* ap = attn + (((size_t)b * HH + h) * NN + i) * NN;
    float acc = 0.f;
    for (int j = 0; j < NN; ++j)
        acc += ap[j] * v[((size_t)b * NN + j) * DD + tid];
    ctx[(size_t)bi * DD + tid] = acc;
}

extern "C" void kernel_launch(void* const* d_in, const int* in_sizes, int n_in,
                              void* d_out, int out_size, void* d_ws, size_t ws_size,
                              hipStream_t stream) {
    (void)in_sizes; (void)n_in; (void)out_size; (void)ws_size;
    const float* desc   = (const float*)d_in[0];
    const float* nv     = (const float*)d_in[1];
    const float* W_gt   = (const float*)d_in[2];
    const float* b_gt   = (const float*)d_in[3];
    const float* topo   = (const float*)d_in[4];
    const float* W_q    = (const float*)d_in[5];
    const float* b_q    = (const float*)d_in[6];
    const float* W_k    = (const float*)d_in[7];
    const float* b_k    = (const float*)d_in[8];
    const float* W_v    = (const float*)d_in[9];
    const float* b_v    = (const float*)d_in[10];
    const float* W_e1   = (const float*)d_in[11];
    const float* b_e1   = (const float*)d_in[12];
    const float* ln_g   = (const float*)d_in[13];
    const float* ln_b   = (const float*)d_in[14];
    const float* W_e2   = (const float*)d_in[15];
    const float* b_e2   = (const float*)d_in[16];
    const float* w_attn = (const float*)d_in[17];
    const float* b_attn = (const float*)d_in[18];
    const float* W_o    = (const float*)d_in[19];
    const float* b_o    = (const float*)d_in[20];

    float* ws   = (float*)d_ws;
    float* dmat = ws;                       // B*S*D
    float* amat = dmat + BB * SS * DD;      // B*S*D
    float* bmat = amat + BB * SS * DD;      // B*S*D
    float* qmat = bmat + BB * SS * DD;      // B*N*D
    float* kmat = qmat + BB * NN * DD;
    float* vmat = kmat + BB * NN * DD;
    float* ctx  = vmat + BB * NN * DD;
    float* na   = ctx  + BB * NN * DD;      // B*N*N
    float* invd = na   + BB * NN * NN;      // B*S
    float* invv = invd + BB * SS;           // B*S
    float* sq   = invv + BB * SS;           // B*N*H
    float* sk   = sq   + BB * NN * HH;      // B*N*H
    float* U    = sk   + BB * NN * HH;      // D*H
    float* c0   = U    + DD * HH;           // H

    float* out  = (float*)d_out;            // B*N*D
    float* attn = out + BB * NN * DD;       // B*H*N*N

    const int Ms   = BB * SS;               // 1024
    const int Mn   = BB * NN;               // 1032
    const int gs   = (Ms / 16) * 16;        // tiles for M=1024
    const int gn   = ((Mn + 15) / 16) * 16; // tiles for M=1032

    // Projections (WMMA f32)
    gemm_wmma<<<gs, 32, 0, stream>>>(desc, W_gt, b_gt, dmat, Ms);
    gemm_wmma<<<gn, 32, 0, stream>>>(nv, W_q, b_q, qmat, Mn);
    gemm_wmma<<<gn, 32, 0, stream>>>(nv, W_k, b_k, kmat, Mn);
    gemm_wmma<<<gn, 32, 0, stream>>>(nv, W_v, b_v, vmat, Mn);
    gemm_wmma<<<gs, 32, 0, stream>>>(dmat, W_e1, nullptr, amat, Ms);
    gemm_wmma<<<gs, 32, 0, stream>>>(dmat, W_e1 + DD * DD, nullptr, bmat, Ms);

    // Norms + hard adjacency (WMMA Gram matrices)
    norms_kernel<<<BB * SS / 8, 256, 0, stream>>>(dmat, nv, invd, invv);
    adj_wmma<<<dim3(64, BB), 32, 0, stream>>>(dmat, nv, invd, invv, topo, na);
    na_boundary<<<(BB * NN + 255) / 256, 256, 0, stream>>>(na);

    // Folded edge projection + per-node score terms
    precompute_u<<<1, 256, 0, stream>>>(W_e2, b_e2, w_attn, U, c0);
    sqsk_kernel<<<(BB * NN * HH + 255) / 256, 256, 0, stream>>>(qmat, kmat, w_attn, sq, sk);

    // Fused LN + head dots + mask -> raw scores in attn region
    scores_kernel<<<BB * NN, 256, 0, stream>>>(amat, bmat, b_e1, ln_g, ln_b,
                                               U, c0, na, sq, sk, b_attn, attn);
    softmax_kernel<<<BB * HH * NN / 8, 256, 0, stream>>>(attn);
    ctx_kernel<<<BB * NN, 256, 0, stream>>>(attn, vmat, ctx);

    // out = ctx @ W_o + b_o (WMMA f32)
    gemm_wmma<<<gn, 32, 0, stream>>>(ctx, W_o, b_o, out, Mn);
}